// LoKrModule_82214263980641
// MI455X (gfx1250) — compile-verified
//
#include <hip/hip_runtime.h>
#include <hip/hip_bf16.h>

typedef float v2f __attribute__((ext_vector_type(2)));
typedef float v4f __attribute__((ext_vector_type(4)));
typedef float v8f __attribute__((ext_vector_type(8)));

#define M_DIM 16384   // B*S = 4*4096
#define N_DIM 4096    // D_OUT
#define K_DIM 4096    // D_IN
#define KT    32      // K slice staged in LDS
#define LDSK  36      // padded LDS row stride (floats): 16B-aligned, conflict-free
#define NKT   (K_DIM / KT)

// ---- CDNA5 async global->LDS copy of one 128x32 f32 x-tile (ASYNCcnt path) ----
__device__ __forceinline__ void async_load_x_tile(const float* __restrict__ x,
                                                  unsigned lds_base, int M0,
                                                  int tid, int kt) {
  const int c4 = (tid & 7) * 4;                       // float4 column in slice
  const unsigned long long sbase =
      (unsigned long long)(x + (size_t)M0 * K_DIM + (size_t)kt * KT);  // uniform
#pragma unroll
  for (int it = 0; it < 4; ++it) {
    const int row = (tid >> 3) + it * 32;             // 0..127
    const unsigned voff = (unsigned)((row * K_DIM + c4) * sizeof(float));
    const unsigned lds  = lds_base + (unsigned)((row * LDSK + c4) * sizeof(float));
    asm volatile("global_load_async_to_lds_b128 %0, %1, %2"
                 :
                 : "v"(lds), "v"(voff), "s"(sbase)
                 : "memory");
  }
}

// Load one 128x32 slice of Weff = W0 + 0.25*w1[i,j]*w2[p,q] into regs (needs VALU fixup).
__device__ __forceinline__ void load_w_tile(const float* __restrict__ W0,
                                            const float* __restrict__ w1,
                                            const float* __restrict__ w2,
                                            int N0, int tid, int kt, v4f wb[4]) {
  const int c4 = (tid & 7) * 4;
  const int gk = kt * KT + c4;
  // Tile fits inside one (i,j) kron block: i = N0>>9, j = gk>>9 are uniform.
  const float w1s = w1[((N0 >> 9) << 3) + (gk >> 9)] * 0.25f;  // MULTIPLIER*SCALE
#pragma unroll
  for (int it = 0; it < 4; ++it) {
    const int row = (tid >> 3) + it * 32;
    const v4f w0v = *(const v4f*)(W0 + (size_t)(N0 + row) * K_DIM + gk);
    const v4f w2v = *(const v4f*)(w2 + (size_t)((N0 + row) & 511) * 512 + (gk & 511));
    wb[it] = w0v + w2v * w1s;                         // fused LoKr branch
  }
}

__device__ __forceinline__ void store_w_tile(float* __restrict__ Bs, int tid,
                                             const v4f wb[4]) {
  const int c4 = (tid & 7) * 4;
#pragma unroll
  for (int it = 0; it < 4; ++it) {
    const int row = (tid >> 3) + it * 32;
    *(v4f*)(Bs + row * LDSK + c4) = wb[it];
  }
}

__global__ __launch_bounds__(256)
void lokr_fused_gemm(const float* __restrict__ x, const float* __restrict__ W0,
                     const float* __restrict__ bias, const float* __restrict__ w1,
                     const float* __restrict__ w2, float* __restrict__ out) {
  __shared__ float As[2][128 * LDSK];   // x tile     (m-major, k inner) — filled by async DMA
  __shared__ float Bs[2][128 * LDSK];   // Weff tile  (n-major, k inner) — filled via VGPRs

  const int tid  = threadIdx.x;
  const int lane = tid & 31;
  const int wave = tid >> 5;
  const int l15  = lane & 15;
  const int hi   = lane >> 4;           // 0: lanes 0-15, 1: lanes 16-31

  const int M0 = blockIdx.y * 128;
  const int N0 = blockIdx.x * 128;
  const int wm = (wave & 1) * 64;       // wave's M offset in tile (2 waves in M)
  const int wn = (wave >> 1) * 32;      // wave's N offset in tile (4 waves in N)

  const unsigned asA0 = (unsigned)(uintptr_t)&As[0][0];
  const unsigned asA1 = (unsigned)(uintptr_t)&As[1][0];

  const v8f vzero = {0.f, 0.f, 0.f, 0.f, 0.f, 0.f, 0.f, 0.f};
  v8f acc[4][2];
#pragma unroll
  for (int mb = 0; mb < 4; ++mb)
#pragma unroll
    for (int nb = 0; nb < 2; ++nb) acc[mb][nb] = vzero;

  v4f wb[4];
  // Prologue: stage tile 0 (x via async DMA, Weff via registers).
  async_load_x_tile(x, asA0, M0, tid, 0);
  load_w_tile(W0, w1, w2, N0, tid, 0, wb);
  store_w_tile(&Bs[0][0], tid, wb);
  asm volatile("s_wait_asynccnt 0x0" ::: "memory");
  __syncthreads();

  int buf = 0;
  for (int kt = 0; kt < NKT; ++kt) {
    const bool has_next = (kt + 1) < NKT;
    if (has_next) {
      // buf^1 is free (all its readers finished before last barrier):
      // overlap async x DMA and W prefetch with the WMMA work below.
      async_load_x_tile(x, (buf ? asA0 : asA1), M0, tid, kt + 1);
      load_w_tile(W0, w1, w2, N0, tid, kt + 1, wb);
    }
    if (kt + 2 < NKT) {  // L2 warm-up two stages ahead -> global_prefetch_b8
      const int gk = (kt + 2) * KT + (tid & 7) * 4;
      __builtin_prefetch(x  + (size_t)(M0 + (tid >> 3)) * K_DIM + gk, 0, 3);
      __builtin_prefetch(W0 + (size_t)(N0 + (tid >> 3)) * K_DIM + gk, 0, 3);
    }

    const float* Ab = &As[buf][0];
    const float* Bb = &Bs[buf][0];
#pragma unroll
    for (int k = 0; k < KT; k += 4) {
      const int kcol = k + hi * 2;      // A/B 16x4: lanes 16-31 hold K=2,3
      v2f afrag[4], bfrag[2];
#pragma unroll
      for (int mb = 0; mb < 4; ++mb)
        afrag[mb] = *(const v2f*)(Ab + (wm + mb * 16 + l15) * LDSK + kcol);
#pragma unroll
      for (int nb = 0; nb < 2; ++nb)
        bfrag[nb] = *(const v2f*)(Bb + (wn + nb * 16 + l15) * LDSK + kcol);
#pragma unroll
      for (int mb = 0; mb < 4; ++mb)
#pragma unroll
        for (int nb = 0; nb < 2; ++nb)
          acc[mb][nb] = __builtin_amdgcn_wmma_f32_16x16x4_f32(
              false, afrag[mb], false, bfrag[nb],
              (short)0, acc[mb][nb], false, false);
    }
    __syncthreads();                     // all reads of buf done
    if (has_next) store_w_tile(&Bs[buf ^ 1][0], tid, wb);
    asm volatile("s_wait_asynccnt 0x0" ::: "memory");  // own DMA into buf^1 done
    __syncthreads();                     // stage buf^1 visible to all waves
    buf ^= 1;
  }

  // Epilogue: C/D layout — VGPR v holds M = v + 8*hi; lanes 0-15 are N=0..15.
#pragma unroll
  for (int mb = 0; mb < 4; ++mb)
#pragma unroll
    for (int nb = 0; nb < 2; ++nb) {
      const int col = N0 + wn + nb * 16 + l15;
      const float bv = bias[col];
#pragma unroll
      for (int v = 0; v < 8; ++v) {
        const int rowm = M0 + wm + mb * 16 + hi * 8 + v;
        out[(size_t)rowm * N_DIM + col] = acc[mb][nb][v] + bv;
      }
    }
}

extern "C" void kernel_launch(void* const* d_in, const int* in_sizes, int n_in,
                              void* d_out, int out_size, void* d_ws, size_t ws_size,
                              hipStream_t stream) {
  const float* x  = (const float*)d_in[0];
  const float* W0 = (const float*)d_in[1];
  const float* b  = (const float*)d_in[2];
  const float* w1 = (const float*)d_in[3];
  const float* w2 = (const float*)d_in[4];
  float* out = (float*)d_out;

  dim3 grid(N_DIM / 128, M_DIM / 128);  // (32, 128)
  lokr_fused_gemm<<<grid, 256, 0, stream>>>(x, W0, b, w1, w2, out);
}